// Tri_MMDLoss_50044958933036
// MI455X (gfx1250) — compile-verified
//
#include <hip/hip_runtime.h>
#include <hip/hip_bf16.h>

typedef __attribute__((ext_vector_type(2))) float v2f;
typedef __attribute__((ext_vector_type(8))) float v8f;

#define BROWS 4096
#define DDIM  128
#define NTOT  12288   // 3 * BROWS

// -------- workspace layout (float offsets) --------
// ws[0]           : sumSq accumulator (all 3 matrices)
// ws[1]           : AP kernel-sum accumulator
// ws[2..129]      : column sums s[k], k = 0..127
// ws[130]         : coef c = -1/(16*bw)
// ws[256..4351]   : sqa[4096]
// ws[4352..8447]  : sqp[4096]
#define WS_SUMSQ  0
#define WS_APACC  1
#define WS_COLSUM 2
#define WS_COEF   130
#define WS_SQA    256
#define WS_SQP    (256 + BROWS)

__global__ void zero_ws_kernel(float* ws) {
    int i = threadIdx.x;
    if (i < 131) ws[i] = 0.0f;  // sumSq, apAcc, colsum[128], coef
}

// One wave (32 lanes) per row: squared L2 norm of anchor/pos rows.
__global__ void row_sq_kernel(const float* __restrict__ A,
                              const float* __restrict__ P,
                              float* __restrict__ sqa,
                              float* __restrict__ sqp) {
    int wave = (blockIdx.x * blockDim.x + threadIdx.x) >> 5;
    int lane = threadIdx.x & 31;
    const float* src;
    float* dst;
    int row;
    if (wave < BROWS) { src = A; dst = sqa; row = wave; }
    else              { src = P; dst = sqp; row = wave - BROWS; }
    float4 v = ((const float4*)(src + (size_t)row * DDIM))[lane];  // 32*4 = 128
    float s = v.x * v.x + v.y * v.y + v.z * v.z + v.w * v.w;
#pragma unroll
    for (int off = 16; off > 0; off >>= 1) s += __shfl_xor(s, off, 32);
    if (lane == 0) dst[row] = s;
}

// Column sums over all 12288 rows + total sum of squares.
// 96 blocks x 128 threads; block b handles rows [128*b, 128*b+128).
__global__ void colsum_kernel(const float* __restrict__ A,
                              const float* __restrict__ P,
                              const float* __restrict__ Ng,
                              float* __restrict__ colsum,
                              float* __restrict__ sumSq) {
    int col = threadIdx.x;             // 0..127
    int rowBase = blockIdx.x * 128;    // multiples of 128; never straddles a source
    const float* src;
    int r0;
    if (rowBase < BROWS)            { src = A;  r0 = rowBase; }
    else if (rowBase < 2 * BROWS)   { src = P;  r0 = rowBase - BROWS; }
    else                            { src = Ng; r0 = rowBase - 2 * BROWS; }
    float cs = 0.0f, ss = 0.0f;
    for (int i = 0; i < 128; ++i) {
        float v = src[(size_t)(r0 + i) * DDIM + col];
        cs += v;
        ss += v * v;
    }
    atomicAdd(&colsum[col], cs);
    __shared__ float red[128];
    red[col] = ss;
    __syncthreads();
#pragma unroll
    for (int off = 64; off > 0; off >>= 1) {
        if (col < off) red[col] += red[col + off];
        __syncthreads();
    }
    if (col == 0) atomicAdd(sumSq, red[0]);
}

// bandwidth from closed form: sum(l2) = 2N*sumSq - 2*||colsum||^2
__global__ void bandwidth_kernel(const float* __restrict__ colsum,
                                 const float* __restrict__ sumSq,
                                 float* __restrict__ coef) {
    if (threadIdx.x == 0 && blockIdx.x == 0) {
        double s2 = 0.0;
        for (int k = 0; k < DDIM; ++k) {
            double c = (double)colsum[k];
            s2 += c * c;
        }
        const double n = (double)NTOT;
        double suml2 = 2.0 * n * (double)sumSq[0] - 2.0 * s2;
        double bw0 = suml2 / (n * n - n);
        double bw  = bw0 / 4.0;                 // / KERNEL_MUL^(KERNEL_NUM//2)
        coef[0] = (float)(-1.0 / (16.0 * bw));  // u = exp(coef * l2)
    }
}

// Anchor x Pos block: one wave computes a 16(M) x 64(N) strip via 4 WMMA
// accumulators (A fragment reused 4x). fp32 WMMA: V_WMMA_F32_16X16X4_F32.
__global__ void __launch_bounds__(256)
ap_block_kernel(const float* __restrict__ A,
                const float* __restrict__ P,
                const float* __restrict__ sqa,
                const float* __restrict__ sqp,
                const float* __restrict__ coef,
                float* __restrict__ apAcc) {
    const int lane = threadIdx.x & 31;
    const int wave = threadIdx.x >> 5;
    const int waveGlobal = blockIdx.x * 8 + wave;   // 16384 waves total
    const int nStripsN = BROWS / 64;                // 64
    const int tm = waveGlobal / nStripsN;           // 0..255
    const int tn = waveGlobal % nStripsN;           // 0..63
    const int row0 = tm * 16;
    const int col0 = tn * 64;

    const int h   = lane >> 4;    // half-wave select (K pair)
    const int l16 = lane & 15;

    // A frag lane(16h+m): v2f = A[row0+m][k+2h .. k+2h+1]
    // B frag lane(16h+n): v2f = P[col0+n][k+2h .. k+2h+1]
    const float* aPtr = A + (size_t)(row0 + l16) * DDIM + 2 * h;
    const float* pPtr = P + (size_t)(col0 + l16) * DDIM + 2 * h;

    v8f acc0 = {}, acc1 = {}, acc2 = {}, acc3 = {};
#pragma unroll 4
    for (int k = 0; k < DDIM; k += 4) {
        v2f af = *(const v2f*)(aPtr + k);
        v2f b0 = *(const v2f*)(pPtr + k);
        v2f b1 = *(const v2f*)(pPtr + 16 * DDIM + k);
        v2f b2 = *(const v2f*)(pPtr + 32 * DDIM + k);
        v2f b3 = *(const v2f*)(pPtr + 48 * DDIM + k);
        acc0 = __builtin_amdgcn_wmma_f32_16x16x4_f32(false, af, false, b0, (short)0, acc0, false, false);
        acc1 = __builtin_amdgcn_wmma_f32_16x16x4_f32(false, af, false, b1, (short)0, acc1, false, false);
        acc2 = __builtin_amdgcn_wmma_f32_16x16x4_f32(false, af, false, b2, (short)0, acc2, false, false);
        acc3 = __builtin_amdgcn_wmma_f32_16x16x4_f32(false, af, false, b3, (short)0, acc3, false, false);
    }

    const float c = coef[0];
    float sa[8];
#pragma unroll
    for (int r = 0; r < 8; ++r) sa[r] = sqa[row0 + r + 8 * h];  // M = r + 8h

    float local = 0.0f;
    // kernels sum = u + u^2 + u^4 + u^8 + u^16, u = exp(c * l2)
#define AP_EPILOG(ACC, F)                                        \
    {                                                            \
        float sp = sqp[col0 + 16 * (F) + l16];                   \
        _Pragma("unroll") for (int r = 0; r < 8; ++r) {          \
            float l2 = sa[r] + sp - 2.0f * (ACC)[r];             \
            float u  = __expf(c * l2);                           \
            float u2 = u * u, u4 = u2 * u2;                      \
            float u8 = u4 * u4, u16 = u8 * u8;                   \
            local += u + u2 + u4 + u8 + u16;                     \
        }                                                        \
    }
    AP_EPILOG(acc0, 0)
    AP_EPILOG(acc1, 1)
    AP_EPILOG(acc2, 2)
    AP_EPILOG(acc3, 3)
#undef AP_EPILOG

#pragma unroll
    for (int off = 16; off > 0; off >>= 1) local += __shfl_xor(local, off, 32);
    if (lane == 0) atomicAdd(apAcc, local);
}

__global__ void finalize_kernel(const float* __restrict__ apAcc, float* __restrict__ out) {
    if (threadIdx.x == 0 && blockIdx.x == 0) {
        const double bb = (double)BROWS * (double)BROWS;
        out[0] = (float)(-2.0 * (double)apAcc[0] / bb);  // -(AP + PA), PA == AP
    }
}

extern "C" void kernel_launch(void* const* d_in, const int* in_sizes, int n_in,
                              void* d_out, int out_size, void* d_ws, size_t ws_size,
                              hipStream_t stream) {
    const float* anchor = (const float*)d_in[0];
    const float* pos    = (const float*)d_in[1];
    const float* neg    = (const float*)d_in[2];
    float* out = (float*)d_out;
    float* ws  = (float*)d_ws;

    float* sumSq  = ws + WS_SUMSQ;
    float* apAcc  = ws + WS_APACC;
    float* colsum = ws + WS_COLSUM;
    float* coef   = ws + WS_COEF;
    float* sqa    = ws + WS_SQA;
    float* sqp    = ws + WS_SQP;

    zero_ws_kernel<<<1, 256, 0, stream>>>(ws);
    // 8192 waves (one per anchor/pos row), 8 waves per block
    row_sq_kernel<<<1024, 256, 0, stream>>>(anchor, pos, sqa, sqp);
    // 12288 rows / 128 rows-per-block
    colsum_kernel<<<96, 128, 0, stream>>>(anchor, pos, neg, colsum, sumSq);
    bandwidth_kernel<<<1, 1, 0, stream>>>(colsum, sumSq, coef);
    // 256 (M strips) * 64 (N strips) = 16384 waves, 8 per block
    ap_block_kernel<<<2048, 256, 0, stream>>>(anchor, pos, sqa, sqp, coef, apAcc);
    finalize_kernel<<<1, 1, 0, stream>>>(apAcc, out);
}